// KWinners_30270929502271
// MI455X (gfx1250) — compile-verified
//
#include <hip/hip_runtime.h>
#include <math.h>

#define NCOLS 4096
#define TPB   256
#define EPT   (NCOLS / TPB)        // 16 scalar elements per thread
#define V4PT  (NCOLS / (TPB * 4))  // 4 float4 per thread

typedef float v4f __attribute__((ext_vector_type(4)));
typedef int   v4i_t __attribute__((vector_size(16)));

__global__ __launch_bounds__(TPB) void kwinners_kernel(
    const float* __restrict__ x,
    const float* __restrict__ duty,
    const int*   __restrict__ kptr,
    const float* __restrict__ bsp,
    float* __restrict__ out)
{
    __shared__ __align__(16) float    s_x[NCOLS];
    __shared__ unsigned               s_key[NCOLS];
    __shared__ int                    s_hist[TPB];
    __shared__ int                    s_cum[TPB];
    __shared__ unsigned               s_prefix;
    __shared__ int                    s_rem;
    __shared__ int                    s_cnt;

    const int    tid  = threadIdx.x;
    const size_t row  = blockIdx.x;
    const float* __restrict__ xrow = x   + row * (size_t)NCOLS;
    float*       __restrict__ orow = out + row * (size_t)NCOLS;

    const int   K  = *kptr;
    const float bs = fmaxf(*bsp, 0.0f);
    const float td = (float)K / (float)NCOLS;

    // ---------- stage row into LDS (CDNA5 async path if available) ----------
#if defined(__gfx1250__) && __has_builtin(__builtin_amdgcn_global_load_async_to_lds_b128)
    {
        v4i_t* gp = (v4i_t*)xrow;   // global source, int4-typed as the builtin expects
        v4i_t* lp = (v4i_t*)s_x;    // LDS destination
        for (int v = 0; v < V4PT; ++v) {
            int e4 = v * TPB + tid;
            __builtin_amdgcn_global_load_async_to_lds_b128(gp + e4, lp + e4, 0, 0);
        }
        asm volatile("s_wait_asynccnt 0" ::: "memory");
    }
#else
    for (int v = 0; v < V4PT; ++v) {
        int e4 = v * TPB + tid;
        v4f val = __builtin_nontemporal_load((const v4f*)xrow + e4);
        ((v4f*)s_x)[e4] = val;
    }
#endif
    __syncthreads();

    // ---------- keys: order-preserving uint32 of boosted values ----------
    for (int v = 0; v < EPT; ++v) {
        int i = v * TPB + tid;
        float boosted = s_x[i] * expf((td - duty[i]) * bs);
        unsigned u = __float_as_uint(boosted);
        s_key[i] = (u & 0x80000000u) ? ~u : (u | 0x80000000u);
    }
    if (tid == 0) { s_prefix = 0u; s_rem = K; }
    __syncthreads();

    // ---------- 4-pass radix select: exact K-th largest key ----------
    for (int shift = 24; shift >= 0; shift -= 8) {
        s_hist[tid] = 0;
        __syncthreads();
        const unsigned pfx = s_prefix;
        const int      rem = s_rem;
        const unsigned maskAbove = (shift == 24) ? 0u : (0xFFFFFFFFu << (shift + 8));

        for (int v = 0; v < EPT; ++v) {
            int i = v * TPB + tid;
            unsigned ky = s_key[i];
            if ((ky & maskAbove) == pfx)
                atomicAdd(&s_hist[(ky >> shift) & 0xFFu], 1);
        }
        __syncthreads();

        // descending inclusive scan: s_cum[t] = sum of hist over bins >= (255 - t)
        s_cum[tid] = s_hist[(TPB - 1) - tid];
        __syncthreads();
        for (int off = 1; off < TPB; off <<= 1) {
            int a = s_cum[tid];
            int b = (tid >= off) ? s_cum[tid - off] : 0;
            __syncthreads();
            s_cum[tid] = a + b;
            __syncthreads();
        }
        int incl = s_cum[tid];
        int excl = incl - s_hist[(TPB - 1) - tid];
        if (excl < rem && rem <= incl) {              // exactly one thread hits
            s_prefix = pfx | ((unsigned)((TPB - 1) - tid) << shift);
            s_rem    = rem - excl;
        }
        __syncthreads();
    }

    const unsigned T = s_prefix;   // K-th largest key
    const int      r = s_rem;      // # elements equal to T to accept (lowest idx first)

    // ---------- binary search for tie cutoff index C ----------
    int lo = 0, hi = NCOLS - 1;
    while (lo < hi) {
        int mid = (lo + hi) >> 1;
        if (tid == 0) s_cnt = 0;
        __syncthreads();
        int c = 0;
        for (int v = 0; v < EPT; ++v) {
            int i = v * TPB + tid;
            c += (s_key[i] == T && i <= mid) ? 1 : 0;
        }
        if (c) atomicAdd(&s_cnt, c);
        __syncthreads();
        int total = s_cnt;
        __syncthreads();
        if (total >= r) hi = mid; else lo = mid + 1;
    }
    const int C = lo;

    // ---------- write output (non-temporal, 128-bit) ----------
    for (int v = 0; v < V4PT; ++v) {
        int e4   = v * TPB + tid;
        int base = e4 * 4;
        v4f o;
        for (int c = 0; c < 4; ++c) {
            int i = base + c;
            unsigned ky = s_key[i];
            bool sel = (ky > T) | ((ky == T) & (i <= C));
            o[c] = sel ? s_x[i] : 0.0f;
        }
        __builtin_nontemporal_store(o, (v4f*)(orow + (size_t)base));
    }
}

extern "C" void kernel_launch(void* const* d_in, const int* in_sizes, int n_in,
                              void* d_out, int out_size, void* d_ws, size_t ws_size,
                              hipStream_t stream) {
    const float* x    = (const float*)d_in[0];
    const float* duty = (const float*)d_in[1];
    const int*   kptr = (const int*)d_in[2];
    const float* bsp  = (const float*)d_in[3];
    float*       out  = (float*)d_out;

    int rows = out_size / NCOLS;   // B = 16384
    kwinners_kernel<<<rows, TPB, 0, stream>>>(x, duty, kptr, bsp, out);
}